// UHGAttention_65438121721901
// MI455X (gfx1250) — compile-verified
//
#include <hip/hip_runtime.h>
#include <math.h>

typedef __attribute__((ext_vector_type(2))) float v2f;
typedef __attribute__((ext_vector_type(8))) float v8f;

#define N_ROWS 4096
#define C_DIM  256
#define TJ     128   // j-tile staged in LDS
#define TPR    8     // threads cooperating on one row
#define RPB    32    // rows per block (256 threads)

// ---------------------------------------------------------------------------
// Kernel 1: qkv = x @ [query;key_w;value;0]^T  via V_WMMA_F32_16X16X4_F32
// One wave per 16-row tile of x. 64 chained f32 WMMAs (K = 256, 4 per step).
// ---------------------------------------------------------------------------
__global__ __launch_bounds__(32) void uhg_qkv_wmma(
    const float* __restrict__ x,
    const float* __restrict__ query,
    const float* __restrict__ key_w,
    const float* __restrict__ value,
    float* __restrict__ qkv)
{
    const int lane = threadIdx.x;      // 0..31
    const int half = lane >> 4;        // 0 or 1
    const int lm   = lane & 15;
    const int base = blockIdx.x * 16;  // first row of this 16-row tile

    const float* xrow = x + (size_t)(base + lm) * C_DIM;

    // B = W^T where W row n: [query(0..3) | key_w(4..7) | value(8..11) | 0 pad]
    const int n = lm;
    const float* wrow;
    bool wvalid = true;
    if (n < 4)       wrow = query + n * C_DIM;
    else if (n < 8)  wrow = key_w + (n - 4) * C_DIM;
    else if (n < 12) wrow = value + (n - 8) * C_DIM;
    else           { wrow = query; wvalid = false; }

    v8f c = {};
    for (int k = 0; k < C_DIM; k += 4) {
        const int kk = k + 2 * half;   // ISA A/B layout: VGPR v holds K = v + 2*half
        v2f a, b;
        a.x = xrow[kk];
        a.y = xrow[kk + 1];
        b.x = wvalid ? wrow[kk]     : 0.0f;
        b.y = wvalid ? wrow[kk + 1] : 0.0f;
        c = __builtin_amdgcn_wmma_f32_16x16x4_f32(
                /*neg_a=*/false, a, /*neg_b=*/false, b,
                /*c_mod=*/(short)0, c, /*reuse_a=*/false, /*reuse_b=*/false);
    }

    // C layout: VGPR r -> M = r + 8*half, N = lane&15
#pragma unroll
    for (int r = 0; r < 8; ++r) {
        const int m = r + 8 * half;
        qkv[(size_t)(base + m) * 16 + lm] = c[r];
    }
}

// ---------------------------------------------------------------------------
// Kernel 2: flash-style online softmax over summed per-head UHG distances.
// 8 threads per row, each handles a strided j-partition; partials merged in
// LDS with a log-sum-exp merge. Writes outH[row][h] = (attention @ V)[row][h].
// ---------------------------------------------------------------------------
__global__ __launch_bounds__(256) void uhg_attn_flash(
    const float* __restrict__ qkv,
    float* __restrict__ outH)
{
    __shared__ float sK[TJ * 4];       // k projections
    __shared__ float sA[TJ * 4];       // 1 - k^2
    __shared__ float sV[TJ * 4];       // V projections
    __shared__ float sM[256 * 6];      // per-thread partials: m, l, acc[4]

    const int t   = threadIdx.x;
    const int sub = t & (TPR - 1);
    const int rl  = t >> 3;
    const int row = blockIdx.x * RPB + rl;

    float qh[4], aq[4];
#pragma unroll
    for (int h = 0; h < 4; ++h) {
        const float v = qkv[(size_t)row * 16 + h];
        qh[h] = v;
        aq[h] = 1.0f - v * v;
    }

    float m = -__builtin_inff();
    float l = 0.0f;
    float acc[4] = {0.f, 0.f, 0.f, 0.f};

    for (int j0 = 0; j0 < N_ROWS; j0 += TJ) {
        // cooperative stage of this j-tile
        for (int idx = t; idx < TJ * 4; idx += 256) {
            const int jl = idx >> 2, h = idx & 3;
            const float kv = qkv[(size_t)(j0 + jl) * 16 + 4 + h];
            sK[idx] = kv;
            sA[idx] = 1.0f - kv * kv;
            sV[idx] = qkv[(size_t)(j0 + jl) * 16 + 8 + h];
        }
        __syncthreads();

        for (int jl = sub; jl < TJ; jl += TPR) {
            float dist = 0.0f;
#pragma unroll
            for (int h = 0; h < 4; ++h) {
                const float diff = qh[h] - sK[jl * 4 + h];
                float den = aq[h] * sA[jl * 4 + h];
                den = fmaxf(den, 1e-8f);
                float arg = fmaf(2.0f * diff * diff,
                                 __builtin_amdgcn_rcpf(den), 1.0f);
                arg = fmaxf(arg, 1.0f + 1e-7f);
                // arccosh(arg) = log(arg + sqrt(arg^2 - 1))
                dist += __logf(arg + __builtin_amdgcn_sqrtf(fmaf(arg, arg, -1.0f)));
            }
            const float score = dist * 0.25f;          // / heads
            const float mn   = fmaxf(m, score);
            const float corr = __expf(m - mn);
            const float p    = __expf(score - mn);
            l = l * corr + p;
#pragma unroll
            for (int h = 0; h < 4; ++h)
                acc[h] = acc[h] * corr + p * sV[jl * 4 + h];
            m = mn;
        }
        __syncthreads();
    }

    // merge the TPR partials of each row (log-sum-exp merge)
    sM[t * 6 + 0] = m;
    sM[t * 6 + 1] = l;
#pragma unroll
    for (int h = 0; h < 4; ++h) sM[t * 6 + 2 + h] = acc[h];
    __syncthreads();

    if (sub == 0) {
        const int b0 = rl * TPR;
        float mstar = -__builtin_inff();
        for (int s = 0; s < TPR; ++s)
            mstar = fmaxf(mstar, sM[(b0 + s) * 6 + 0]);
        float L = 0.0f, A[4] = {0.f, 0.f, 0.f, 0.f};
        for (int s = 0; s < TPR; ++s) {
            const float w = __expf(sM[(b0 + s) * 6 + 0] - mstar);
            L += w * sM[(b0 + s) * 6 + 1];
#pragma unroll
            for (int h = 0; h < 4; ++h)
                A[h] += w * sM[(b0 + s) * 6 + 2 + h];
        }
        const float rL = __builtin_amdgcn_rcpf(L);
#pragma unroll
        for (int h = 0; h < 4; ++h)
            outH[(size_t)row * 4 + h] = A[h] * rL;
    }
}

// ---------------------------------------------------------------------------
// Kernel 3: out = ((attnV @ value) / H) then project_to_uhg row-normalization.
// One block per row; 256-thread LDS reduction for the row norm.
// ---------------------------------------------------------------------------
__global__ __launch_bounds__(256) void uhg_out_project(
    const float* __restrict__ outH,
    const float* __restrict__ value,
    float* __restrict__ out)
{
    __shared__ float red[256];
    __shared__ float vh[4];
    const int t   = threadIdx.x;
    const int row = blockIdx.x;

    if (t < 4) vh[t] = outH[(size_t)row * 4 + t];
    __syncthreads();

    float val = 0.25f * (vh[0] * value[0 * C_DIM + t] +
                         vh[1] * value[1 * C_DIM + t] +
                         vh[2] * value[2 * C_DIM + t] +
                         vh[3] * value[3 * C_DIM + t]);

    red[t] = val * val;
    __syncthreads();
    for (int s = 128; s > 0; s >>= 1) {
        if (t < s) red[t] += red[t + s];
        __syncthreads();
    }
    const float norm = __builtin_amdgcn_sqrtf(red[0]);

    float o;
    if (norm < 1e-8f)
        o = (t == 0 ? 1.0f : 0.0f) / (1.0f + 1e-8f);  // canonical, renormalized
    else
        o = val / (norm + 1e-8f);

    out[(size_t)row * C_DIM + t] = o;
}

// ---------------------------------------------------------------------------
extern "C" void kernel_launch(void* const* d_in, const int* in_sizes, int n_in,
                              void* d_out, int out_size, void* d_ws, size_t ws_size,
                              hipStream_t stream) {
    (void)in_sizes; (void)n_in; (void)out_size; (void)ws_size;
    const float* x     = (const float*)d_in[0];
    const float* query = (const float*)d_in[1];
    const float* key_w = (const float*)d_in[2];
    const float* value = (const float*)d_in[3];
    float* out = (float*)d_out;

    float* qkv  = (float*)d_ws;               // 4096 * 16 floats (256 KB)
    float* outH = qkv + (size_t)N_ROWS * 16;  // 4096 * 4 floats  (64 KB)

    uhg_qkv_wmma<<<N_ROWS / 16, 32, 0, stream>>>(x, query, key_w, value, qkv);
    uhg_attn_flash<<<N_ROWS / RPB, 256, 0, stream>>>(qkv, outH);
    uhg_out_project<<<N_ROWS, 256, 0, stream>>>(outH, value, out);
}